// InputExpander_33801392619791
// MI455X (gfx1250) — compile-verified
//
#include <hip/hip_runtime.h>

#define NN    10000   // N_NODES
#define NF    128     // N_FEATS
#define BSZ_  32
#define DD    100000  // D

// ---------------------------------------------------------------------------
// Kernel 1: zero-fill the 163.84MB output.
// gfx1250 path: async store-from-LDS (ASYNCcnt-tracked DMA-style stores,
// 16B/lane from a shared LDS zero block). Falls back to plain float4 stores
// if the builtin isn't declared by this toolchain.
// ---------------------------------------------------------------------------
#if defined(__HIP_DEVICE_COMPILE__) && \
    __has_builtin(__builtin_amdgcn_global_store_async_from_lds_b128)
#define USE_ASYNC_FILL 1
#else
#define USE_ASYNC_FILL 0
#endif

typedef int v4i __attribute__((vector_size(16)));
typedef __attribute__((address_space(1))) v4i gas_v4i;  // global (AS1) 16B vec
typedef __attribute__((address_space(3))) v4i lds_v4i;  // LDS    (AS3) 16B vec

__global__ void ie_fill_zero(float4* __restrict__ out4, long long n4,
                             float* __restrict__ out, long long n) {
  long long stride = (long long)gridDim.x * blockDim.x;
  long long i = (long long)blockIdx.x * blockDim.x + threadIdx.x;
#if USE_ASYNC_FILL
  __shared__ __align__(16) float zbuf[4];
  if (threadIdx.x == 0) {
    zbuf[0] = 0.0f; zbuf[1] = 0.0f; zbuf[2] = 0.0f; zbuf[3] = 0.0f;
  }
  __syncthreads();
  lds_v4i* lsrc = (lds_v4i*)&zbuf[0];
  for (; i < n4; i += stride) {
    gas_v4i* g = (gas_v4i*)(out4 + i);
    // (global dst, lds src, imm offset, imm cpol) ; tracked by ASYNCcnt
    __builtin_amdgcn_global_store_async_from_lds_b128(g, lsrc, 0, 0);
  }
#if __has_builtin(__builtin_amdgcn_s_wait_asynccnt)
  __builtin_amdgcn_s_wait_asynccnt(0);
#endif
#else
  const float4 z = make_float4(0.0f, 0.0f, 0.0f, 0.0f);
  for (; i < n4; i += stride) out4[i] = z;
#endif
  // Scalar tail (out_size not divisible by 4 — not expected here, but cheap).
  if (blockIdx.x == 0 && threadIdx.x == 0) {
    for (long long t = n4 * 4; t < n; ++t) out[t] = 0.0f;
  }
}

// ---------------------------------------------------------------------------
// Kernel 2: the scatter. One lane per (b, j) element.
// Inputs are streamed exactly once -> non-temporal loads so they don't evict
// the zero-filled output lines from the 192MB L2 (output must stay resident
// so the random 4B stores merge in L2 and each line hits HBM once).
// ---------------------------------------------------------------------------
__global__ void ie_scatter(const float* __restrict__ vals,
                           const int* __restrict__ nodes,
                           const int* __restrict__ feats,
                           float* __restrict__ out) {
  int j = blockIdx.x * blockDim.x + threadIdx.x;
  if (j >= DD) return;
  int b = blockIdx.y;
  int s = b * DD + j;                       // < 3.2M, fits int
  int   node = __builtin_nontemporal_load(nodes + s);
  int   feat = __builtin_nontemporal_load(feats + s);
  float v    = __builtin_nontemporal_load(vals + s);
  // dst < 32*10000*128 = 40.96M, fits unsigned; *128 is a shift.
  unsigned dst = ((unsigned)b * NN + (unsigned)node) * NF + (unsigned)feat;
  out[dst] = v;                             // RT store: merge in L2
}

extern "C" void kernel_launch(void* const* d_in, const int* in_sizes, int n_in,
                              void* d_out, int out_size, void* d_ws, size_t ws_size,
                              hipStream_t stream) {
  (void)in_sizes; (void)n_in; (void)d_ws; (void)ws_size;
  const float* vals  = (const float*)d_in[0];  // flat_inputs [32,100000] f32
  const int*   nodes = (const int*)d_in[1];    // node_ids    [32,100000]
  const int*   feats = (const int*)d_in[2];    // feat_ids    [32,100000]
  float*       out   = (float*)d_out;          // [32,10000,128] f32

  long long n  = (long long)out_size;          // 40,960,000
  long long n4 = n >> 2;                       // 10,240,000 float4 stores

  // Fill: grid-stride, 8192 blocks x 256 thr (2.1M lanes, ~5 iters of 16B).
  ie_fill_zero<<<8192, 256, 0, stream>>>((float4*)out, n4, out, n);

  // Scatter: grid (ceil(100000/256), 32 batches).
  dim3 g((DD + 255) / 256, BSZ_);
  ie_scatter<<<g, 256, 0, stream>>>(vals, nodes, feats, out);
}